// GINConv_layer_72688026518088
// MI455X (gfx1250) — compile-verified
//
#include <hip/hip_runtime.h>

#define N_NODES    100000
#define N_EDGES    1600000
#define HID        128
#define NUM_GRAPHS 128
#define LN_EPS     1e-5f
#define GN_EPS     1e-5f

typedef __attribute__((ext_vector_type(2))) float v2f;
typedef __attribute__((ext_vector_type(8))) float v8f;

__device__ __forceinline__ float atomicAddF(float* p, float v) {
  // Lowered to global_atomic_add_f32 (no CAS loop) on gfx1250.
  return unsafeAtomicAdd(p, v);
}

// ---------------------------------------------------------------------------
// h0 = (1 + eps) * node
// ---------------------------------------------------------------------------
__global__ void k_init_h(const float4* __restrict__ node,
                         const float* __restrict__ eps,
                         float4* __restrict__ h) {
  int i = blockIdx.x * blockDim.x + threadIdx.x;
  const int total = N_NODES * HID / 4;
  if (i >= total) return;
  float s = 1.0f + eps[0];
  float4 v = node[i];
  v.x *= s; v.y *= s; v.z *= s; v.w *= s;
  h[i] = v;
}

// ---------------------------------------------------------------------------
// h0[dst] += node[src]  (GIN neighbor-sum; L2-resident atomics)
// One thread = one edge x one float4 feature chunk.
// ---------------------------------------------------------------------------
__global__ void k_edge_scatter(const float* __restrict__ node,
                               const int* __restrict__ ei,
                               float* __restrict__ h) {
  long long i = (long long)blockIdx.x * blockDim.x + threadIdx.x;
  const long long total = (long long)N_EDGES * (HID / 4);
  if (i >= total) return;
  int e = (int)(i >> 5);
  int q = (((int)i) & 31) * 4;
  int src = ei[e];
  int dst = ei[N_EDGES + e];
  const float4 v = *(const float4*)(node + (long long)src * HID + q);
  float* o = h + (long long)dst * HID + q;
  atomicAddF(o + 0, v.x);
  atomicAddF(o + 1, v.y);
  atomicAddF(o + 2, v.z);
  atomicAddF(o + 3, v.w);
}

// ---------------------------------------------------------------------------
// Y = [relu(LN(X @ W + b))]  (DO_LN=true)  or  Y = X @ W + b  (DO_LN=false)
// Block: 256 threads = 8 waves; 16 rows x 128 cols per block.
// Wave w computes the 16x16 tile at columns [16w, 16w+16) with
// v_wmma_f32_16x16x4_f32, K = 128 in 32 steps.
// ---------------------------------------------------------------------------
template <bool DO_LN>
__global__ __launch_bounds__(256) void k_gemm(const float* __restrict__ X,
                                              const float* __restrict__ W,
                                              const float* __restrict__ bias,
                                              const float* __restrict__ gamma,
                                              const float* __restrict__ beta,
                                              float* __restrict__ Y) {
  __shared__ float Ash[16][HID + 4];   // +4 pad: conflict-free column reads
  __shared__ float Csh[16][HID + 4];

  const int tid = threadIdx.x;
  const int rowBase = blockIdx.x * 16;

  // Cooperative A-tile load: 16 rows x 128 cols, float4 per thread-iter.
  for (int i = tid; i < 16 * (HID / 4); i += 256) {
    int r = i >> 5;            // HID/4 == 32
    int c = (i & 31) * 4;
    *(float4*)&Ash[r][c] =
        *(const float4*)(X + (long long)(rowBase + r) * HID + c);
  }
  __syncthreads();

  const int wave = tid >> 5;
  const int lane = tid & 31;
  const int half = lane >> 4;   // 0: lanes 0-15, 1: lanes 16-31
  const int l    = lane & 15;
  const int colBase = wave * 16;

  v8f acc = {};
  for (int kb = 0; kb < HID; kb += 4) {
    const int k0 = kb + half * 2;
    v2f a, b;
    // A fragment (16x4, MxK): lane l holds M=l; VGPR0/1 = K=k0/k0+1
    a[0] = Ash[l][k0];
    a[1] = Ash[l][k0 + 1];
    // B fragment (4x16, KxN): lane l holds N=colBase+l; rows K striped
    b[0] = W[(long long)k0 * HID + colBase + l];
    b[1] = W[(long long)(k0 + 1) * HID + colBase + l];
    acc = __builtin_amdgcn_wmma_f32_16x16x4_f32(
        /*neg_a=*/false, a, /*neg_b=*/false, b,
        /*c_mod=*/(short)0, acc, /*reuse_a=*/false, /*reuse_b=*/false);
  }

  const float bcol = bias[colBase + l];

  if (DO_LN) {
    // Stage C tile (+bias) to LDS: VGPR r holds M = r + 8*half, N = l.
    #pragma unroll
    for (int r = 0; r < 8; ++r)
      Csh[r + half * 8][colBase + l] = acc[r] + bcol;
    __syncthreads();

    // Fused LayerNorm + ReLU: wave w handles rows 2w, 2w+1.
    #pragma unroll
    for (int rr = 0; rr < 2; ++rr) {
      const int row = wave * 2 + rr;
      float x[4];
      float s = 0.0f, ss = 0.0f;
      #pragma unroll
      for (int j = 0; j < 4; ++j) {
        x[j] = Csh[row][lane + 32 * j];
        s  += x[j];
        ss += x[j] * x[j];
      }
      #pragma unroll
      for (int m = 16; m >= 1; m >>= 1) {
        s  += __shfl_xor(s,  m, 32);
        ss += __shfl_xor(ss, m, 32);
      }
      const float mu  = s * (1.0f / HID);
      const float var = ss * (1.0f / HID) - mu * mu;
      const float inv = rsqrtf(var + LN_EPS);
      float* yrow = Y + (long long)(rowBase + row) * HID;
      #pragma unroll
      for (int j = 0; j < 4; ++j) {
        const int c = lane + 32 * j;
        const float yv = (x[j] - mu) * inv * gamma[c] + beta[c];
        yrow[c] = fmaxf(yv, 0.0f);
      }
    }
  } else {
    #pragma unroll
    for (int r = 0; r < 8; ++r) {
      const int m = r + half * 8;
      Y[(long long)(rowBase + m) * HID + colBase + l] = acc[r] + bcol;
    }
  }
}

// ---------------------------------------------------------------------------
// GraphNorm pipeline
// ---------------------------------------------------------------------------
__global__ void k_zero(float* __restrict__ p, int n) {
  int i = blockIdx.x * blockDim.x + threadIdx.x;
  if (i < n) p[i] = 0.0f;
}

__global__ void k_gsum(const float* __restrict__ h, const int* __restrict__ gid,
                       float* __restrict__ gsum, float* __restrict__ gcount) {
  long long i = (long long)blockIdx.x * blockDim.x + threadIdx.x;
  const long long total = (long long)N_NODES * 32;
  if (i >= total) return;
  int n = (int)(i >> 5);
  int q = (((int)i) & 31) * 4;
  int g = gid[n];
  const float4 v = *(const float4*)(h + (long long)n * HID + q);
  float* o = gsum + g * HID + q;
  atomicAddF(o + 0, v.x);
  atomicAddF(o + 1, v.y);
  atomicAddF(o + 2, v.z);
  atomicAddF(o + 3, v.w);
  if (q == 0) atomicAddF(gcount + g, 1.0f);
}

// gmean[g][f] = gn_mean_scale[f] * sum[g][f] / max(count[g], 1)
__global__ void k_mean(const float* __restrict__ gsum,
                       const float* __restrict__ gcount,
                       const float* __restrict__ mscale,
                       float* __restrict__ gmean) {
  int i = blockIdx.x * blockDim.x + threadIdx.x;
  if (i >= NUM_GRAPHS * HID) return;
  int g = i >> 7;
  int f = i & (HID - 1);
  float c = fmaxf(gcount[g], 1.0f);
  gmean[i] = mscale[f] * gsum[i] / c;
}

__global__ void k_gvar(const float* __restrict__ h, const int* __restrict__ gid,
                       const float* __restrict__ gmean, float* __restrict__ gvar) {
  long long i = (long long)blockIdx.x * blockDim.x + threadIdx.x;
  const long long total = (long long)N_NODES * 32;
  if (i >= total) return;
  int n = (int)(i >> 5);
  int q = (((int)i) & 31) * 4;
  int g = gid[n];
  const float4 v = *(const float4*)(h + (long long)n * HID + q);
  const float* m = gmean + g * HID + q;
  float* o = gvar + g * HID + q;
  float s0 = v.x - m[0], s1 = v.y - m[1], s2 = v.z - m[2], s3 = v.w - m[3];
  atomicAddF(o + 0, s0 * s0);
  atomicAddF(o + 1, s1 * s1);
  atomicAddF(o + 2, s2 * s2);
  atomicAddF(o + 3, s3 * s3);
}

// gvar[g][f] <- rsqrt(var[g][f]/count + eps)
__global__ void k_varfin(float* __restrict__ gvar,
                         const float* __restrict__ gcount) {
  int i = blockIdx.x * blockDim.x + threadIdx.x;
  if (i >= NUM_GRAPHS * HID) return;
  int g = i >> 7;
  float c = fmaxf(gcount[g], 1.0f);
  gvar[i] = rsqrtf(gvar[i] / c + GN_EPS);
}

__global__ void k_out(const float* __restrict__ h, const int* __restrict__ gid,
                      const float* __restrict__ gmean,
                      const float* __restrict__ gvinv,
                      const float* __restrict__ w, const float* __restrict__ b,
                      float* __restrict__ out) {
  long long i = (long long)blockIdx.x * blockDim.x + threadIdx.x;
  const long long total = (long long)N_NODES * 32;
  if (i >= total) return;
  int n = (int)(i >> 5);
  int q = (((int)i) & 31) * 4;
  int g = gid[n];
  const float4 v = *(const float4*)(h + (long long)n * HID + q);
  const float* m  = gmean + g * HID + q;
  const float* iv = gvinv + g * HID + q;
  float4 r;
  r.x = fmaxf(w[q + 0] * (v.x - m[0]) * iv[0] + b[q + 0], 0.0f);
  r.y = fmaxf(w[q + 1] * (v.y - m[1]) * iv[1] + b[q + 1], 0.0f);
  r.z = fmaxf(w[q + 2] * (v.z - m[2]) * iv[2] + b[q + 2], 0.0f);
  r.w = fmaxf(w[q + 3] * (v.w - m[3]) * iv[3] + b[q + 3], 0.0f);
  *(float4*)(out + (long long)n * HID + q) = r;
}

// ---------------------------------------------------------------------------
extern "C" void kernel_launch(void* const* d_in, const int* in_sizes, int n_in,
                              void* d_out, int out_size, void* d_ws,
                              size_t ws_size, hipStream_t stream) {
  const float* node  = (const float*)d_in[0];
  const int*   ei    = (const int*)d_in[1];
  // d_in[2] edge_attr: unused by reference
  const int*   gid   = (const int*)d_in[3];   // per-node graph id (sorted)
  const float* eps   = (const float*)d_in[4];
  const float* W1    = (const float*)d_in[5];
  const float* b1    = (const float*)d_in[6];
  const float* g1    = (const float*)d_in[7];
  const float* be1   = (const float*)d_in[8];
  const float* W2    = (const float*)d_in[9];
  const float* b2    = (const float*)d_in[10];
  const float* g2    = (const float*)d_in[11];
  const float* be2   = (const float*)d_in[12];
  const float* W3    = (const float*)d_in[13];
  const float* b3    = (const float*)d_in[14];
  const float* gnw   = (const float*)d_in[15];
  const float* gnb   = (const float*)d_in[16];
  const float* gnms  = (const float*)d_in[17];

  float* ws    = (float*)d_ws;
  float* bufA  = ws;                                   // [N, H]
  float* bufB  = bufA + (size_t)N_NODES * HID;         // [N, H]
  float* gsum  = bufB + (size_t)N_NODES * HID;         // [G, H]
  float* gmean = gsum + NUM_GRAPHS * HID;              // [G, H]
  float* gvar  = gmean + NUM_GRAPHS * HID;             // [G, H]
  float* gcnt  = gvar + NUM_GRAPHS * HID;              // [G]

  const int nh4    = N_NODES * HID / 4;
  const long long edgeThreads = (long long)N_EDGES * 32;
  const long long nodeThreads = (long long)N_NODES * 32;
  const int gemmBlocks = N_NODES / 16;                 // 100000 % 16 == 0

  // h0 = (1+eps)*node, then scatter-add neighbor rows (L2-resident atomics)
  k_init_h<<<(nh4 + 255) / 256, 256, 0, stream>>>((const float4*)node, eps,
                                                  (float4*)bufA);
  k_edge_scatter<<<(int)((edgeThreads + 255) / 256), 256, 0, stream>>>(node, ei,
                                                                       bufA);
  // MLP: Lin->LN->ReLU, Lin->LN->ReLU, Lin (WMMA + fused LN/ReLU)
  k_gemm<true><<<gemmBlocks, 256, 0, stream>>>(bufA, W1, b1, g1, be1, bufB);
  k_gemm<true><<<gemmBlocks, 256, 0, stream>>>(bufB, W2, b2, g2, be2, bufA);
  k_gemm<false><<<gemmBlocks, 256, 0, stream>>>(bufA, W3, b3, nullptr, nullptr,
                                                bufB);
  // GraphNorm
  const int accN = NUM_GRAPHS * HID * 3 + NUM_GRAPHS;  // gsum,gmean,gvar,gcnt
  k_zero<<<(accN + 255) / 256, 256, 0, stream>>>(gsum, accN);
  k_gsum<<<(int)((nodeThreads + 255) / 256), 256, 0, stream>>>(bufB, gid, gsum,
                                                               gcnt);
  k_mean<<<(NUM_GRAPHS * HID + 255) / 256, 256, 0, stream>>>(gsum, gcnt, gnms,
                                                             gmean);
  k_gvar<<<(int)((nodeThreads + 255) / 256), 256, 0, stream>>>(bufB, gid, gmean,
                                                               gvar);
  k_varfin<<<(NUM_GRAPHS * HID + 255) / 256, 256, 0, stream>>>(gvar, gcnt);
  k_out<<<(int)((nodeThreads + 255) / 256), 256, 0, stream>>>(
      bufB, gid, gmean, gvar, gnw, gnb, (float*)d_out);
}